// DCModuleOptimized_67748814127111
// MI455X (gfx1250) — compile-verified
//
#include <hip/hip_runtime.h>
#include <cstdint>

#define BATCH 32
#define H 512
#define W 512
#define NP 255          // (512-3)/2 + 1 patches per dim
#define TP 16           // patches per tile dim -> 16x16 tile grid
#define TIN 33          // max input rows/cols staged per tile (2*16+1)
#define SSEC 1104       // float stride between LDS image sections (>=33*33, 64B-mult)

typedef uint32_t v4u __attribute__((ext_vector_type(4)));
typedef uint32_t v8u __attribute__((ext_vector_type(8)));

// CDNA5 Tensor Data Mover: DMA a trows x tcols fp32 tile (global row stride W)
// into LDS at lds_byte, packed contiguously (row stride = tcols floats).
// D# groups built per CDNA5 ISA; 2-D tensor -> groups 2/3 NULL.
// Tracked by TENSORcnt. EXEC is ignored by TDM; issue from one wave only.
__device__ __forceinline__ void tdm_load_tile(uint32_t lds_byte, const float* gptr,
                                              uint32_t tcols, uint32_t trows) {
  const uint64_t ga = (uint64_t)(uintptr_t)gptr;
  const uint32_t TD = 1u << 30;   // huge tensor dims: tiles pre-clamped in-bounds,
                                  // so OOB handling never engages
  v4u g0;
  g0.x = 1u;                                   // count=1, is_restore=0, gather off
  g0.y = lds_byte;                             // lds_addr (bytes)
  g0.z = (uint32_t)ga;                         // global_addr[31:0]
  g0.w = (uint32_t)(ga >> 32) | (2u << 30);    // global_addr[56:32] | type=2

  v8u g1;
  g1.s0 = 2u << 16;                            // workgroup_mask=0, data_size=4B
  g1.s1 = (TD & 0xffffu) << 16;                // atomic_barrier=0 | tensor_dim0[15:0]
  g1.s2 = ((TD >> 16) & 0xffffu)               // tensor_dim0[31:16]
        | ((TD & 0xffffu) << 16);              // tensor_dim1[15:0]
  g1.s3 = ((TD >> 16) & 0xffffu)               // tensor_dim1[31:16]
        | (tcols << 16);                       // tile_dim0
  g1.s4 = trows;                               // tile_dim1 | tile_dim2=0
  g1.s5 = (uint32_t)W;                         // tensor_dim0_stride[31:0] = 512
  g1.s6 = 0u;                                  // stride0 hi | tensor_dim1_stride lo
  g1.s7 = 0u;                                  // tensor_dim1_stride hi (unused, 2-D)

  asm volatile("tensor_load_to_lds %0, %1"
               :: "s"(g0), "s"(g1)
               : "memory");
}

__global__ __launch_bounds__(256)
void dc_pool_kernel(const float* __restrict__ anchor,
                    const float* __restrict__ posi,
                    const float* __restrict__ nega,
                    float* __restrict__ outP,
                    float* __restrict__ outN)
{
  __shared__ float lds[3 * SSEC];

  const int tid = threadIdx.x;
  const int b   = blockIdx.z;
  const int py0 = blockIdx.y * TP;
  const int px0 = blockIdx.x * TP;
  const int r0  = py0 * 2;
  const int c0  = px0 * 2;
  const size_t ib = (size_t)b * (size_t)(H * W);

  // Clamped tile geometry: edge tiles drop the last row/col, which only the
  // nonexistent patch row/col 255 would have read.
  const int tcols = min(TIN, W - c0);   // 33 interior, 32 on tx==15
  const int trows = min(TIN, H - r0);   // 33 interior, 32 on ty==15
  const size_t goff = ib + (size_t)r0 * W + c0;

  if (tid < 32) {   // wave 0 drives the TDM; EXEC is ignored by tensor ops
    const uint32_t l0 = (uint32_t)(uintptr_t)(&lds[0]);
    tdm_load_tile(l0,                 anchor + goff, (uint32_t)tcols, (uint32_t)trows);
    tdm_load_tile(l0 + 4u * SSEC,     posi   + goff, (uint32_t)tcols, (uint32_t)trows);
    tdm_load_tile(l0 + 8u * SSEC,     nega   + goff, (uint32_t)tcols, (uint32_t)trows);
    __builtin_amdgcn_s_wait_tensorcnt(0);   // this wave's DMA has landed in LDS
  }
  __syncthreads();                          // visible to all waves

  const int lpy = tid >> 4;
  const int lpx = tid & 15;
  const int py = py0 + lpy;
  const int px = px0 + lpx;
  if (py >= NP || px >= NP) return;   // idle threads in the ragged edge tiles

  // ---- 3x3 argmin/argmax of |a - c|, selecting c; strict compares keep
  //      jnp's first-occurrence tie semantics (scan in k = dy*3+dx order) ----
  const int base = (2 * lpy) * tcols + 2 * lpx;
  float mnDp, mxDp, mnVp, mxVp, mnDn, mxDn, mnVn, mxVn;
  #pragma unroll
  for (int dy = 0; dy < 3; ++dy) {
    #pragma unroll
    for (int dx = 0; dx < 3; ++dx) {
      const int o = base + dy * tcols + dx;
      const float a = lds[o];
      const float p = lds[o + SSEC];
      const float n = lds[o + 2 * SSEC];
      const float dp = fabsf(a - p);
      const float dn = fabsf(a - n);
      if (dy == 0 && dx == 0) {
        mnDp = mxDp = dp; mnVp = mxVp = p;
        mnDn = mxDn = dn; mnVn = mxVn = n;
      } else {
        if (dp < mnDp) { mnDp = dp; mnVp = p; }
        if (dp > mxDp) { mxDp = dp; mxVp = p; }
        if (dn < mnDn) { mnDn = dn; mnVn = n; }
        if (dn > mxDn) { mxDn = dn; mxVn = n; }
      }
    }
  }
  const float sp = mnVp + mxVp;
  const float sn = mnVn + mxVn;

  // ---- Scatter to full resolution. Patch (py,px) owns pixels {2py,2py+1} x
  //      {2px,2px+1}; edge patches (index 254) also own row/col 510 (valued)
  //      and row/col 511 (masked to zero). float2 stores, 8B aligned. ----
  float* OP = outP + ib;
  float* ON = outN + ib;
  const int y0 = 2 * py, x0 = 2 * px;
  const float2 vp = make_float2(sp, sp);
  const float2 vn = make_float2(sn, sn);
  const float2 zz = make_float2(0.f, 0.f);

  *(float2*)(OP + (size_t)y0 * W + x0)       = vp;
  *(float2*)(OP + (size_t)(y0 + 1) * W + x0) = vp;
  *(float2*)(ON + (size_t)y0 * W + x0)       = vn;
  *(float2*)(ON + (size_t)(y0 + 1) * W + x0) = vn;

  if (px == NP - 1) {  // columns 510 (value) and 511 (zero)
    const float2 ep = make_float2(sp, 0.f);
    const float2 en = make_float2(sn, 0.f);
    *(float2*)(OP + (size_t)y0 * W + (W - 2))       = ep;
    *(float2*)(OP + (size_t)(y0 + 1) * W + (W - 2)) = ep;
    *(float2*)(ON + (size_t)y0 * W + (W - 2))       = en;
    *(float2*)(ON + (size_t)(y0 + 1) * W + (W - 2)) = en;
  }
  if (py == NP - 1) {  // rows 510 (value) and 511 (zero)
    *(float2*)(OP + (size_t)(H - 2) * W + x0) = vp;
    *(float2*)(OP + (size_t)(H - 1) * W + x0) = zz;
    *(float2*)(ON + (size_t)(H - 2) * W + x0) = vn;
    *(float2*)(ON + (size_t)(H - 1) * W + x0) = zz;
    if (px == NP - 1) {  // 2x2 corner
      const float2 ep = make_float2(sp, 0.f);
      const float2 en = make_float2(sn, 0.f);
      *(float2*)(OP + (size_t)(H - 2) * W + (W - 2)) = ep;
      *(float2*)(OP + (size_t)(H - 1) * W + (W - 2)) = zz;
      *(float2*)(ON + (size_t)(H - 2) * W + (W - 2)) = en;
      *(float2*)(ON + (size_t)(H - 1) * W + (W - 2)) = zz;
    }
  }
}

extern "C" void kernel_launch(void* const* d_in, const int* in_sizes, int n_in,
                              void* d_out, int out_size, void* d_ws, size_t ws_size,
                              hipStream_t stream) {
  (void)in_sizes; (void)n_in; (void)out_size; (void)d_ws; (void)ws_size;
  const float* anchor = (const float*)d_in[0];
  const float* posi   = (const float*)d_in[1];
  const float* nega   = (const float*)d_in[2];
  float* out  = (float*)d_out;
  float* outP = out;                                   // min_pos + max_pos
  float* outN = out + (size_t)BATCH * H * W;           // min_neg + max_neg

  dim3 grid((NP + TP - 1) / TP, (NP + TP - 1) / TP, BATCH);  // 16 x 16 x 32
  dc_pool_kernel<<<grid, 256, 0, stream>>>(anchor, posi, nega, outP, outN);
}